// LocalPoolNet_34291018891286
// MI455X (gfx1250) — compile-verified
//
#include <hip/hip_runtime.h>
#include <math.h>

// ---------------- problem constants ----------------
#define NB    100        // graphs
#define NPG0  500        // nodes per graph (level 0)
#define NEDGE 600000     // total edges (length constant across levels; emask shrinks)
#define EPG   6000       // edges per graph (edge e belongs to graph e/EPG, all levels)
#define HDIM  128        // feature dim (F_IN == H == 128)
#define TILE  125        // dst-node tile per block for LDS aggregation (125*128*4 = 64000 B)

typedef float v2f __attribute__((ext_vector_type(2)));
typedef float v8f __attribute__((ext_vector_type(8)));

// ---------------- edge init ----------------
__global__ void init_edges_kernel(const int* __restrict__ eidx,
                                  int* __restrict__ esrc, int* __restrict__ edst,
                                  float* __restrict__ emask) {
    int e = blockIdx.x * 256 + threadIdx.x;
    if (e >= NEDGE) return;
    esrc[e] = eidx[e];
    edst[e] = eidx[NEDGE + e];
    emask[e] = 1.0f;
}

// ---------------- fused SAGE layer: LDS aggregation + WMMA GEMM + bias + ReLU -------------
// One block per (graph, 125-node dst tile). Phase 1: scan the graph's 6000 edges,
// ds_add_f32-accumulate msg rows for dst nodes in this tile (LDS atomics, not L2 RMW).
// Phase 2: out = relu( (agg/max(cnt,1)) @ Wl + bl + x @ Wr ) via v_wmma_f32_16x16x4_f32,
// A-operand streamed from LDS, weights/x from global (L2-resident).
__global__ __launch_bounds__(1024) void sage_fused_kernel(
        const float* __restrict__ x,
        const int* __restrict__ esrc, const int* __restrict__ edst,
        const float* __restrict__ emask,
        const float* __restrict__ Wl, const float* __restrict__ blv,
        const float* __restrict__ Wr,
        float* __restrict__ out, int n_per, int ntiles) {
    __shared__ float sAgg[TILE * HDIM];   // 64000 B
    __shared__ float sCnt[TILE];

    const int tid = threadIdx.x;
    const int b   = blockIdx.x / ntiles;
    const int d0  = (blockIdx.x % ntiles) * TILE;       // local dst base of this tile
    const int gbase = b * n_per;                        // global node base of this graph

    // ---- zero LDS ----
    for (int i = tid; i < TILE * HDIM; i += 1024) sAgg[i] = 0.0f;
    for (int i = tid; i < TILE; i += 1024) sCnt[i] = 0.0f;
    __syncthreads();

    // ---- phase 1: edge scan, 32 edge-slots x 32 lanes, float4 per lane ----
    {
        const int slot = tid >> 5;          // 0..31
        const int lane = tid & 31;
        const int f0   = lane * 4;          // 4 features per lane -> 128
        for (int e0 = 0; e0 < EPG; e0 += 32) {
            int el = e0 + slot;
            if (el < EPG) {
                int e = b * EPG + el;
                float m = emask[e];
                if (m != 0.0f) {
                    int dl = edst[e] - gbase - d0;      // local dst within tile
                    if (dl >= 0 && dl < TILE) {
                        int s = esrc[e];
                        const float4 xv = *(const float4*)(x + (size_t)s * HDIM + f0);
                        atomicAdd(&sAgg[dl * HDIM + f0 + 0], xv.x * m);
                        atomicAdd(&sAgg[dl * HDIM + f0 + 1], xv.y * m);
                        atomicAdd(&sAgg[dl * HDIM + f0 + 2], xv.z * m);
                        atomicAdd(&sAgg[dl * HDIM + f0 + 3], xv.w * m);
                        if (lane == 0) atomicAdd(&sCnt[dl], m);
                    }
                }
            }
        }
    }
    __syncthreads();

    // ---- phase 2: WMMA GEMM over this tile (128 padded rows x 128 cols) ----
    // 8 M-tiles x 8 N-tiles = 64 wave-tiles; 32 waves -> 2 tiles each.
    {
        const int wave = tid >> 5;
        const int lane = tid & 31;
        const int half = lane >> 4;     // 0: K pair {k,k+1}; 1: K pair {k+2,k+3}
        const int l15  = lane & 15;

        for (int wt = wave; wt < 64; wt += 32) {
            const int mt = wt >> 3;
            const int n0 = (wt & 7) * 16;
            int ml = mt * 16 + l15;
            int mlc = ml < TILE ? ml : TILE - 1;        // clamp reads; stores predicated
            const float rinv = 1.0f / fmaxf(sCnt[mlc], 1.0f);
            const float* aggRow = sAgg + mlc * HDIM;
            const float* xRow   = x + (size_t)(gbase + d0 + mlc) * HDIM;

            v8f c;
            #pragma unroll
            for (int r = 0; r < 8; ++r) c[r] = 0.0f;

            for (int k = 0; k < HDIM; k += 4) {
                const int kk = k + half * 2;
                v2f am, ax, bwl, bwr;
                am.x = aggRow[kk] * rinv;  am.y = aggRow[kk + 1] * rinv;   // ds_load from LDS
                ax.x = xRow[kk];           ax.y = xRow[kk + 1];
                bwl.x = Wl[(size_t)kk * HDIM + n0 + l15];
                bwl.y = Wl[(size_t)(kk + 1) * HDIM + n0 + l15];
                bwr.x = Wr[(size_t)kk * HDIM + n0 + l15];
                bwr.y = Wr[(size_t)(kk + 1) * HDIM + n0 + l15];
                c = __builtin_amdgcn_wmma_f32_16x16x4_f32(false, am, false, bwl, (short)0, c, false, false);
                c = __builtin_amdgcn_wmma_f32_16x16x4_f32(false, ax, false, bwr, (short)0, c, false, false);
            }

            const float bias = blv[n0 + l15];
            #pragma unroll
            for (int r = 0; r < 8; ++r) {
                int rowl = mt * 16 + r + half * 8;      // C layout: VGPR r -> M=r / M=r+8
                if (rowl < TILE) {
                    float v = c[r] + bias;
                    out[(size_t)(gbase + d0 + rowl) * HDIM + n0 + l15] = v > 0.0f ? v : 0.0f;
                }
            }
        }
    }
}

// ---------------- ||w|| ----------------
__global__ void norm_kernel(const float* __restrict__ w, float* __restrict__ outNorm) {
    __shared__ float red[128];
    int t = threadIdx.x;
    float v = w[t];
    red[t] = v * v;
    __syncthreads();
    for (int s = 64; s > 0; s >>= 1) { if (t < s) red[t] += red[t + s]; __syncthreads(); }
    if (t == 0) *outNorm = sqrtf(red[0]);
}

// ---------------- score = tanh((x.w)/||w||), one wave per node ----------------
__global__ void score_kernel(const float* __restrict__ x, const float* __restrict__ w,
                             const float* __restrict__ normp,
                             float* __restrict__ score, int nodes) {
    int lane = threadIdx.x & 31;
    int node = blockIdx.x * 8 + (threadIdx.x >> 5);
    if (node >= nodes) return;                 // wave-uniform exit
    const float4* xr = (const float4*)(x + (size_t)node * HDIM);
    const float4* wr = (const float4*)w;
    float4 a = xr[lane], b = wr[lane];
    float s = a.x * b.x + a.y * b.y + a.z * b.z + a.w * b.w;
    #pragma unroll
    for (int o = 16; o > 0; o >>= 1) s += __shfl_xor(s, o, 32);
    if (lane == 0) score[node] = tanhf(s / (*normp));
}

// ---------------- TopK per graph: bitonic sort 512 in LDS ----------------
// rank order: higher score first; tie -> lower index first (matches lax.top_k)
__global__ void topk_kernel(const float* __restrict__ xin, const float* __restrict__ score,
                            float* __restrict__ xout, int* __restrict__ mapping,
                            int n_per, int kkeep) {
    __shared__ float ss[512];
    __shared__ int   si[512];
    const int b = blockIdx.x, tid = threadIdx.x;
    for (int i = tid; i < 512; i += 256) {
        ss[i] = (i < n_per) ? score[b * n_per + i] : -INFINITY;
        si[i] = i;
    }
    __syncthreads();
    for (int kk = 2; kk <= 512; kk <<= 1) {
        for (int jj = kk >> 1; jj > 0; jj >>= 1) {
            for (int i = tid; i < 512; i += 256) {
                int ixj = i ^ jj;
                if (ixj > i) {
                    float s1 = ss[i], s2 = ss[ixj];
                    int i1 = si[i], i2 = si[ixj];
                    bool before12 = (s1 > s2) || (s1 == s2 && i1 < i2);
                    bool asc = ((i & kk) == 0);
                    bool doswap = asc ? !before12 : before12;
                    if (doswap) { ss[i] = s2; ss[ixj] = s1; si[i] = i2; si[ixj] = i1; }
                }
            }
            __syncthreads();
        }
    }
    // gather pooled features scaled by top value
    for (int t = tid; t < kkeep * HDIM; t += 256) {
        int j = t >> 7, f = t & 127;
        int oldl = si[j];
        xout[((size_t)(b * kkeep + j)) * HDIM + f] =
            xin[((size_t)(b * n_per + oldl)) * HDIM + f] * ss[j];
    }
    // node mapping old -> new (-1 = dropped)
    for (int t = tid; t < n_per; t += 256) mapping[b * n_per + t] = -1;
    __syncthreads();
    for (int j = tid; j < kkeep; j += 256) mapping[b * n_per + si[j]] = b * kkeep + j;
}

// ---------------- edge remap ----------------
__global__ void remap_edges_kernel(int* __restrict__ esrc, int* __restrict__ edst,
                                   float* __restrict__ emask, const int* __restrict__ mapping) {
    int e = blockIdx.x * 256 + threadIdx.x;
    if (e >= NEDGE) return;
    int ns = mapping[esrc[e]];
    int nd = mapping[edst[e]];
    float m = emask[e];
    if (ns < 0 || nd < 0) m = 0.0f;
    emask[e] = m;
    esrc[e] = ns < 0 ? 0 : ns;
    edst[e] = nd < 0 ? 0 : nd;
}

// ---------------- readout: z[b] += [max_j x, mean_j x] ----------------
__global__ void readout_kernel(const float* __restrict__ x, float* __restrict__ z, int kkeep) {
    int b = blockIdx.x, f = threadIdx.x;   // 128 threads
    float mx = -INFINITY, sm = 0.0f;
    for (int j = 0; j < kkeep; ++j) {
        float v = x[((size_t)b * kkeep + j) * HDIM + f];
        mx = fmaxf(mx, v);
        sm += v;
    }
    z[b * 256 + f]       += mx;
    z[b * 256 + 128 + f] += sm / (float)kkeep;
}

// ---------------- MLP head ----------------
__global__ void mlp1_kernel(const float* __restrict__ z, const float* __restrict__ W,
                            const float* __restrict__ bvec, float* __restrict__ out) {
    int b = blockIdx.x, o = threadIdx.x;   // 128
    float s = bvec[o];
    for (int i = 0; i < 256; ++i) s += z[b * 256 + i] * W[i * 128 + o];
    out[b * 128 + o] = s > 0.0f ? s : 0.0f;
}
__global__ void mlp2_kernel(const float* __restrict__ z, const float* __restrict__ W,
                            const float* __restrict__ bvec, float* __restrict__ out) {
    int b = blockIdx.x, o = threadIdx.x;   // 64
    float s = bvec[o];
    for (int i = 0; i < 128; ++i) s += z[b * 128 + i] * W[i * 64 + o];
    out[b * 64 + o] = s > 0.0f ? s : 0.0f;
}
__global__ void mlp3_kernel(const float* __restrict__ z, const float* __restrict__ W,
                            const float* __restrict__ bvec, float* __restrict__ out) {
    __shared__ float lg[16];
    int b = blockIdx.x, o = threadIdx.x;   // 32 threads
    if (o < 10) {
        float s = bvec[o];
        for (int i = 0; i < 64; ++i) s += z[b * 64 + i] * W[i * 10 + o];
        lg[o] = s;
    }
    __syncthreads();
    if (o < 10) {
        float m = -INFINITY;
        for (int i = 0; i < 10; ++i) m = fmaxf(m, lg[i]);
        float se = 0.0f;
        for (int i = 0; i < 10; ++i) se += expf(lg[i] - m);
        out[b * 10 + o] = lg[o] - m - logf(se);
    }
}

// ---------------- host orchestration ----------------
static inline char* carve(char*& p, size_t bytes) {
    char* r = p;
    p += (bytes + 255) & ~(size_t)255;
    return r;
}

extern "C" void kernel_launch(void* const* d_in, const int* in_sizes, int n_in,
                              void* d_out, int out_size, void* d_ws, size_t ws_size,
                              hipStream_t stream) {
    (void)in_sizes; (void)n_in; (void)out_size; (void)ws_size;

    const float* x0   = (const float*)d_in[0];
    const int*   eidx = (const int*)d_in[1];
    const float* Wl[3]  = {(const float*)d_in[2], (const float*)d_in[5], (const float*)d_in[8]};
    const float* blv[3] = {(const float*)d_in[3], (const float*)d_in[6], (const float*)d_in[9]};
    const float* Wr[3]  = {(const float*)d_in[4], (const float*)d_in[7], (const float*)d_in[10]};
    const float* pw[3]  = {(const float*)d_in[11], (const float*)d_in[12], (const float*)d_in[13]};
    const float* W1 = (const float*)d_in[14]; const float* b1 = (const float*)d_in[15];
    const float* W2 = (const float*)d_in[16]; const float* b2 = (const float*)d_in[17];
    const float* W3 = (const float*)d_in[18]; const float* b3 = (const float*)d_in[19];

    char* p = (char*)d_ws;
    const size_t NMAX = (size_t)NB * NPG0;                 // 50000
    float* bufA   = (float*)carve(p, NMAX * HDIM * 4);     // conv output h
    float* bufB   = (float*)carve(p, NMAX * HDIM * 4);     // pooled x
    int*   esrc   = (int*)  carve(p, (size_t)NEDGE * 4);
    int*   edst   = (int*)  carve(p, (size_t)NEDGE * 4);
    float* emask  = (float*)carve(p, (size_t)NEDGE * 4);
    int*   mapping= (int*)  carve(p, NMAX * 4);
    float* scores = (float*)carve(p, NMAX * 4);
    float* zbuf   = (float*)carve(p, (size_t)NB * 256 * 4);
    float* m1     = (float*)carve(p, (size_t)NB * 128 * 4);
    float* m2     = (float*)carve(p, (size_t)NB * 64 * 4);
    float* wnorm  = (float*)carve(p, 256);

    init_edges_kernel<<<(NEDGE + 255) / 256, 256, 0, stream>>>(eidx, esrc, edst, emask);
    hipMemsetAsync(zbuf, 0, (size_t)NB * 256 * 4, stream);

    const int nper[3]  = {500, 250, 125};
    const int kkeep[3] = {250, 125, 63};
    int nodes = NB * NPG0;
    const float* xin = x0;

    for (int l = 0; l < 3; ++l) {
        const int ntiles = nper[l] / TILE;     // 4, 2, 1

        sage_fused_kernel<<<NB * ntiles, 1024, 0, stream>>>(
            xin, esrc, edst, emask, Wl[l], blv[l], Wr[l], bufA, nper[l], ntiles);

        norm_kernel<<<1, 128, 0, stream>>>(pw[l], wnorm);
        score_kernel<<<(nodes + 7) / 8, 256, 0, stream>>>(bufA, pw[l], wnorm, scores, nodes);

        topk_kernel<<<NB, 256, 0, stream>>>(bufA, scores, bufB, mapping, nper[l], kkeep[l]);
        remap_edges_kernel<<<(NEDGE + 255) / 256, 256, 0, stream>>>(esrc, edst, emask, mapping);
        readout_kernel<<<NB, 128, 0, stream>>>(bufB, zbuf, kkeep[l]);

        nodes = NB * kkeep[l];
        xin = bufB;
    }

    mlp1_kernel<<<NB, 128, 0, stream>>>(zbuf, W1, b1, m1);
    mlp2_kernel<<<NB, 64, 0, stream>>>(m1, W2, b2, m2);
    mlp3_kernel<<<NB, 32, 0, stream>>>(m2, W3, b3, (float*)d_out);
}